// Model_78915729096710
// MI455X (gfx1250) — compile-verified
//
#include <hip/hip_runtime.h>
#include <math.h>

typedef __attribute__((ext_vector_type(2))) float v2f;
typedef __attribute__((ext_vector_type(8))) float v8f;

#define IS_   512
#define K_    16

// One wave computes a 16(batch) x 16(neuron) output tile via block-diagonal
// f32 WMMA: D[b,n] = sum_{n',k} Xg[b,(n',k)] * (w[n',k] * (n==n')).
// Gathers are branch-free (clamped address + v_cndmask); neuron ids are
// forced into SGPRs via readfirstlane so idx/weight loads use scalar bases.
__global__ __launch_bounds__(128) void spn_wmma_kernel(
    const float* __restrict__ x,             // (64, 512) row-major
    const float* __restrict__ weights,       // (N, 16)
    const float* __restrict__ bias,          // (N,)
    const int*   __restrict__ input_indices, // (N, 16), values in [0, 513)
    const int*   __restrict__ act_id,        // (N,)
    const int*   __restrict__ out_sel,       // (N_OUT,) tape indices = 513 + neuron
    float*       __restrict__ out,           // (64, N_OUT) row-major
    int n_out)
{
    const int lane = threadIdx.x & 31;
    const int bt   = threadIdx.x >> 5;     // batch tile 0..3 (one wave each)
    const int nt   = blockIdx.x;           // neuron tile

    const int col   = lane & 15;           // A row (batch) == D col (neuron) index
    const int khalf = (lane >> 4) << 1;    // K base for this half-wave: 0 or 2
    const int boff  = (bt * 16 + col) * IS_;  // 32-bit element offset into x

    // Preload the 16 neuron ids; readfirstlane pins them (and the derived
    // idx/weight base pointers) to SGPRs -> saddr-form vector loads below.
    int ngs[16];
    #pragma unroll
    for (int np = 0; np < 16; ++np) {
        const int j  = nt * 16 + np;
        const int jc = (j < n_out) ? j : (n_out - 1);
        ngs[np] = __builtin_amdgcn_readfirstlane(out_sel[jc]) - (1 + IS_);
    }

    v8f acc = {};

    #pragma unroll
    for (int np = 0; np < 16; ++np) {      // neuron-within-tile (diagonal block)
        const bool valid = (nt * 16 + np) < n_out;
        const int  ng    = ngs[np];        // wave-uniform, in SGPR
        const int*   __restrict__ idxp = input_indices + ng * K_;
        const float* __restrict__ wp   = weights       + ng * K_;
        const bool mine = valid && (col == np);

        #pragma unroll
        for (int c = 0; c < 4; ++c) {      // K=16 consumed 4 at a time
            const int kk0 = c * 4 + khalf;

            // Contiguous pair loads from scalar bases (b64-pairable).
            const int   i0 = idxp[kk0];
            const int   i1 = idxp[kk0 + 1];
            const float w0 = wp[kk0];
            const float w1 = wp[kk0 + 1];

            // Clamped-address gather + value select (v_max_i32 + v_cndmask).
            int t0 = i0 - 1; t0 = (t0 < 0) ? 0 : t0;
            int t1 = i1 - 1; t1 = (t1 < 0) ? 0 : t1;
            const float xv0 = x[boff + t0];
            const float xv1 = x[boff + t1];

            v2f A;
            A.x = (i0 != 0) ? xv0 : 0.0f;
            A.y = (i1 != 0) ? xv1 : 0.0f;

            v2f Bm;
            Bm.x = mine ? w0 : 0.0f;
            Bm.y = mine ? w1 : 0.0f;

            acc = __builtin_amdgcn_wmma_f32_16x16x4_f32(
                /*neg_a=*/false, A, /*neg_b=*/false, Bm,
                /*c_mod=*/(short)0, acc, /*reuse_a=*/false, /*reuse_b=*/false);
        }
    }

    // Epilogue: bias + activation + store. Lane's D column is `col` (per-lane,
    // so these stay vector loads).
    const int j = nt * 16 + col;
    if (j < n_out) {
        const int   ng = out_sel[j] - (1 + IS_);
        const float bz = bias[ng];
        const int   a  = act_id[ng];
        const int   mbase = bt * 16 + ((lane >> 4) << 3);  // D row base
        #pragma unroll
        for (int v = 0; v < 8; ++v) {
            float h = acc[v] + bz;
            h = (a == 0) ? fmaxf(h, 0.0f) : tanhf(h);
            out[(mbase + v) * n_out + j] = h;
        }
    }
}

extern "C" void kernel_launch(void* const* d_in, const int* in_sizes, int n_in,
                              void* d_out, int out_size, void* d_ws, size_t ws_size,
                              hipStream_t stream) {
    // setup_inputs() order:
    // 0: x (B*IS f32), 1: weights (N*K f32), 2: bias (N f32),
    // 3: input_indices (N*K i32), 4: output_indices (N i32, unused),
    // 5: act_id (N i32), 6: out_sel (N_OUT i32)
    const float* x             = (const float*)d_in[0];
    const float* weights       = (const float*)d_in[1];
    const float* bias          = (const float*)d_in[2];
    const int*   input_indices = (const int*)d_in[3];
    const int*   act_id        = (const int*)d_in[5];
    const int*   out_sel       = (const int*)d_in[6];
    float*       out           = (float*)d_out;

    const int n_out  = in_sizes[6];          // 1000
    const int ntiles = (n_out + 15) / 16;    // 63

    spn_wmma_kernel<<<ntiles, 128, 0, stream>>>(
        x, weights, bias, input_indices, act_id, out_sel, out, n_out);
}